// PhaseClassifierBatch_28389733827069
// MI455X (gfx1250) — compile-verified
//
#include <hip/hip_runtime.h>
#include <math.h>

// ---------------------------------------------------------------------------
// PhaseClassifierBatch for MI455X (gfx1250, wave32).
// Memory-bound: ~0.97 GB of fp32 weights streamed once -> ~42us floor at
// 23.3 TB/s.  Matrix op: V_WMMA_F32_16X16X4_F32 (the only f32 WMMA shape).
// One workgroup per sample keeps activations resident in LDS.
// 1024 threads (32 waves) per block: one 16-row tile per wave per layer,
// maximizing outstanding global_load_b64 clauses for HBM latency hiding.
// ---------------------------------------------------------------------------

typedef __attribute__((ext_vector_type(2))) float v2f;
typedef __attribute__((ext_vector_type(8))) float v8f;

#define DD       512
#define NBINS    314
#define NTHREADS 1024
#define NWAVES   (NTHREADS / 32)
#define BATCH    256

__device__ __forceinline__ float gelu_exact(float v) {
    // exact GELU: 0.5*x*(1+erf(x/sqrt(2)))
    return 0.5f * v * (1.0f + erff(v * 0.70710678118654752440f));
}

// Accumulate y[tile rows] += W_tile * h using V_WMMA_F32_16X16X4_F32.
// A (16x4 f32): lane l<16  -> VGPR0=W[l][k+0],    VGPR1=W[l][k+1]
//               lane l>=16 -> VGPR0=W[l-16][k+2], VGPR1=W[l-16][k+3]
// B (4x16 f32): all 16 columns identical = h[k..k+3] (same per-lane pattern).
// D (16x16):    every column equals y; lanes 0-15 hold y[0..7] in d[0..7],
//               lanes 16-31 hold y[8..15].
__device__ __forceinline__ v8f wmma_matvec_tile(const float* __restrict__ Wrow,
                                                const float* __restrict__ hsl) {
    v8f acc = {0.f, 0.f, 0.f, 0.f, 0.f, 0.f, 0.f, 0.f};
#pragma unroll 8
    for (int k = 0; k < DD; k += 4) {
        v2f a  = *(const v2f*)(Wrow + k);   // global b64, streams the weights
        v2f hb = *(const v2f*)(hsl + k);    // ds b64, broadcast vector slice
        acc = __builtin_amdgcn_wmma_f32_16x16x4_f32(
            /*neg_a=*/false, a, /*neg_b=*/false, hb,
            /*c_mod=*/(short)0, acc, /*reuse_a=*/false, /*reuse_b=*/false);
    }
    return acc;
}

__device__ __forceinline__ void fc_layer(const float* __restrict__ Wb,
                                         const float* __restrict__ bias_b,
                                         const float* __restrict__ hin,
                                         float* __restrict__ hout,
                                         int lane, int wave) {
    const int row15 = lane & 15;
    const int hsel  = (lane >> 4) << 1;          // 0 or 2
    const float* hsl = hin + hsel;
    for (int tile = wave; tile < DD / 16; tile += NWAVES) {  // 1 tile per wave
        const float* Wrow = Wb + (size_t)(tile * 16 + row15) * DD + hsel;
        v8f acc = wmma_matvec_tile(Wrow, hsl);
        if (row15 == 0) {                        // lanes 0 and 16 write 8 each
            int mbase = tile * 16 + ((lane >> 4) << 3);
#pragma unroll
            for (int r = 0; r < 8; ++r) {
                float v = acc[r] + bias_b[mbase + r];
                hout[mbase + r] = gelu_exact(v);
            }
        }
    }
}

__global__ __launch_bounds__(NTHREADS, 1)
void phase_classifier_kernel(const float* __restrict__ x,
                             const float* __restrict__ ln_w,
                             const float* __restrict__ ln_b,
                             const float* __restrict__ fc1_w,
                             const float* __restrict__ fc1_b,
                             const float* __restrict__ fc2_w,
                             const float* __restrict__ fc2_b,
                             const float* __restrict__ fc3_w,
                             const float* __restrict__ fc3_b,
                             const float* __restrict__ out_w,
                             const float* __restrict__ out_b,
                             float* __restrict__ out) {
    __shared__ __align__(16) float hA[DD];
    __shared__ __align__(16) float hB[DD];
    __shared__ __align__(16) float lg[320];
    __shared__ float redA[NWAVES], redB[NWAVES], redC[NWAVES];
    __shared__ float stat[2];

    const int b    = blockIdx.x;
    const int t    = threadIdx.x;
    const int lane = t & 31;
    const int wave = t >> 5;

    // ---------------- LayerNorm (biased var, eps=1e-5) ----------------
    float xv = (t < DD) ? x[(size_t)b * DD + t] : 0.f;   // zeros don't perturb sums
    {
        float s = xv, q = xv * xv;
#pragma unroll
        for (int off = 16; off > 0; off >>= 1) {
            s += __shfl_xor(s, off, 32);
            q += __shfl_xor(q, off, 32);
        }
        if (lane == 0) { redA[wave] = s; redB[wave] = q; }
    }
    __syncthreads();
    if (t == 0) {
        float S = 0.f, Q = 0.f;
        for (int i = 0; i < NWAVES; ++i) { S += redA[i]; Q += redB[i]; }
        float mean = S * (1.0f / DD);
        float var  = Q * (1.0f / DD) - mean * mean;
        stat[0] = mean;
        stat[1] = rsqrtf(var + 1e-5f);
    }
    __syncthreads();
    if (t < DD) {
        hA[t] = (xv - stat[0]) * stat[1] * ln_w[(size_t)b * DD + t] +
                ln_b[(size_t)b * DD + t];
    }
    __syncthreads();

    // ---------------- 3x fc (512x512 matvec + GELU) ----------------
    const size_t woff = (size_t)b * DD * DD;
    fc_layer(fc1_w + woff, fc1_b + (size_t)b * DD, hA, hB, lane, wave);
    __syncthreads();
    fc_layer(fc2_w + woff, fc2_b + (size_t)b * DD, hB, hA, lane, wave);
    __syncthreads();
    fc_layer(fc3_w + woff, fc3_b + (size_t)b * DD, hA, hB, lane, wave);
    __syncthreads();

    // ---------------- output layer: 314 rows in 20 tiles ----------------
    {
        const int row15 = lane & 15;
        const int hsel  = (lane >> 4) << 1;
        const float* hsl = hB + hsel;
        const float* Wb  = out_w + (size_t)b * NBINS * DD;
        for (int tile = wave; tile < 20; tile += NWAVES) {
            int row = tile * 16 + row15;
            if (row > NBINS - 1) row = NBINS - 1;   // clamp: no OOB reads
            const float* Wrow = Wb + (size_t)row * DD + hsel;
            v8f acc = wmma_matvec_tile(Wrow, hsl);
            if (row15 == 0) {
                int mbase = tile * 16 + ((lane >> 4) << 3);
#pragma unroll
                for (int r = 0; r < 8; ++r) {
                    int m = mbase + r;
                    if (m < NBINS)
                        lg[m] = acc[r] + out_b[(size_t)b * NBINS + m];
                }
            }
        }
    }
    __syncthreads();

    // ---------------- softmax (T=1) + circular mean ----------------
    float li = (t < NBINS) ? lg[t] : -3.402823466e38f;
    {
        float mx = li;
#pragma unroll
        for (int off = 16; off > 0; off >>= 1)
            mx = fmaxf(mx, __shfl_xor(mx, off, 32));
        if (lane == 0) redA[wave] = mx;
    }
    __syncthreads();
    if (t == 0) {
        float M = redA[0];
        for (int i = 1; i < NWAVES; ++i) M = fmaxf(M, redA[i]);
        stat[0] = M;
    }
    __syncthreads();

    float e = (t < NBINS) ? expf(li - stat[0]) : 0.f;
    float ctr = -3.14159265358979323846f +
                6.28318530717958647692f * ((float)t + 0.5f) / (float)NBINS;
    float es = e * sinf(ctr);
    float ec = e * cosf(ctr);
    {
#pragma unroll
        for (int off = 16; off > 0; off >>= 1) {
            es += __shfl_xor(es, off, 32);
            ec += __shfl_xor(ec, off, 32);
        }
        if (lane == 0) { redB[wave] = es; redC[wave] = ec; }
    }
    __syncthreads();
    if (t == 0) {
        float S = 0.f, C = 0.f;
        for (int i = 0; i < NWAVES; ++i) { S += redB[i]; C += redC[i]; }
        // atan2 is invariant to the (positive) softmax normalizer
        out[b] = atan2f(S, C);
    }
    if (t < NBINS)
        out[BATCH + (size_t)b * NBINS + t] = lg[t];
}

extern "C" void kernel_launch(void* const* d_in, const int* in_sizes, int n_in,
                              void* d_out, int out_size, void* d_ws, size_t ws_size,
                              hipStream_t stream) {
    (void)in_sizes; (void)n_in; (void)d_ws; (void)ws_size; (void)out_size;
    const float* x     = (const float*)d_in[0];
    const float* ln_w  = (const float*)d_in[1];
    const float* ln_b  = (const float*)d_in[2];
    const float* fc1_w = (const float*)d_in[3];
    const float* fc1_b = (const float*)d_in[4];
    const float* fc2_w = (const float*)d_in[5];
    const float* fc2_b = (const float*)d_in[6];
    const float* fc3_w = (const float*)d_in[7];
    const float* fc3_b = (const float*)d_in[8];
    const float* out_w = (const float*)d_in[9];
    const float* out_b = (const float*)d_in[10];
    float* out = (float*)d_out;

    phase_classifier_kernel<<<BATCH, NTHREADS, 0, stream>>>(
        x, ln_w, ln_b, fc1_w, fc1_b, fc2_w, fc2_b, fc3_w, fc3_b,
        out_w, out_b, out);
}